// HeatLoss_OldGen_3_44032004718828
// MI455X (gfx1250) — compile-verified
//
#include <hip/hip_runtime.h>

// Problem constants from the reference: B=32, C=17, H=W=256.
#define NBATCH 32
#define NCH    17
#define HW     65536          // 256*256 pixels per (batch, channel) plane
#define Q4     (HW / 4)       // float4 groups per plane = 16384
#define NGROUPS (NBATCH * Q4) // 524288 (batch, float4-group) work items
#define MAIN_BLOCKS 2048
#define BLOCK_THREADS 256

typedef __attribute__((ext_vector_type(2))) float v2f;
typedef __attribute__((ext_vector_type(8))) float v8f;

// Full-wave (32 lane) sum using V_WMMA_F32_16X16X4_F32.
// A: VGPR0 = per-lane partial, VGPR1 = 0  -> the 64 A slots hold the 32
// partials once each (rest zeros). B = all-ones 4x16. D[m,n] = row-sum of A,
// identical in every column. Per-lane sum of the 8 D VGPRs covers rows 0-7
// (lanes 0-15) or rows 8-15 (lanes 16-31); one shfl_xor(16) completes the sum
// and broadcasts it to every lane. EXEC must be all ones at every call site.
__device__ __forceinline__ float wave_sum_wmma(float x) {
    v2f a; a[0] = x;    a[1] = 0.0f;
    v2f b; b[0] = 1.0f; b[1] = 1.0f;
    v8f c = {};
    v8f d = __builtin_amdgcn_wmma_f32_16x16x4_f32(
        /*neg_a=*/false, a, /*neg_b=*/false, b,
        /*c_mod=*/(short)0, c, /*reuse_a=*/false, /*reuse_b=*/false);
    float t = ((d[0] + d[1]) + (d[2] + d[3])) + ((d[4] + d[5]) + (d[6] + d[7]));
    t += __shfl_xor(t, 16, 32);
    return t;
}

// Reduce a float4 across a 256-thread block. Result valid in wave 0 (all lanes).
__device__ __forceinline__ float4 block_reduce4(float4 v) {
    __shared__ float4 red[BLOCK_THREADS / 32];
    v.x = wave_sum_wmma(v.x);
    v.y = wave_sum_wmma(v.y);
    v.z = wave_sum_wmma(v.z);
    v.w = wave_sum_wmma(v.w);
    const int wave = threadIdx.x >> 5;
    const int lane = threadIdx.x & 31;
    if (lane == 0) red[wave] = v;
    __syncthreads();
    if (wave == 0) {
        const int nwaves = blockDim.x >> 5;
        float4 u = (lane < nwaves) ? red[lane] : make_float4(0.f, 0.f, 0.f, 0.f);
        u.x = wave_sum_wmma(u.x);
        u.y = wave_sum_wmma(u.y);
        u.z = wave_sum_wmma(u.z);
        u.w = wave_sum_wmma(u.w);
        v = u;
    }
    return v;
}

// Pass 1: streaming masked-L1 partial sums. One float4 group = 4 adjacent
// pixels; the thread loops the 17 channels so the per-pixel "any channel
// positive" mask (m2) is computed in-register with a single read of target.
__global__ void __launch_bounds__(BLOCK_THREADS)
heatloss_partial(const float* __restrict__ inp,
                 const float* __restrict__ tgt,
                 float4* __restrict__ partials) {
    const int tid    = blockIdx.x * blockDim.x + threadIdx.x;
    const int stride = gridDim.x * blockDim.x;

    float s1 = 0.0f;   // sum(m1 * diff)
    float s2 = 0.0f;   // sum over pixels of any * (sum_c diff)
    int   c1 = 0;      // count(m1)
    int   c2 = 0;      // count of any-positive pixels

    for (int g = tid; g < NGROUPS; g += stride) {
        const int b  = g >> 14;        // g / Q4
        const int q4 = g & (Q4 - 1);   // g % Q4
        const int base = b * (NCH * HW) + (q4 << 2);

        float ds0 = 0.f, ds1 = 0.f, ds2 = 0.f, ds3 = 0.f;
        int any0 = 0, any1 = 0, any2 = 0, any3 = 0;

#pragma unroll
        for (int ch = 0; ch < NCH; ++ch) {
            const int off = base + ch * HW;
            const float4 t = *(const float4*)(tgt + off);
            const float4 x = *(const float4*)(inp + off);

            const float d0 = fabsf(x.x - t.x);
            const float d1 = fabsf(x.y - t.y);
            const float d2 = fabsf(x.z - t.z);
            const float d3 = fabsf(x.w - t.w);

            const int m0 = t.x > 0.0f;
            const int m1 = t.y > 0.0f;
            const int m2 = t.z > 0.0f;
            const int m3 = t.w > 0.0f;

            ds0 += d0; ds1 += d1; ds2 += d2; ds3 += d3;
            s1 += m0 ? d0 : 0.0f;
            s1 += m1 ? d1 : 0.0f;
            s1 += m2 ? d2 : 0.0f;
            s1 += m3 ? d3 : 0.0f;
            c1 += m0 + m1 + m2 + m3;
            any0 |= m0; any1 |= m1; any2 |= m2; any3 |= m3;
        }
        s2 += any0 ? ds0 : 0.0f;
        s2 += any1 ? ds1 : 0.0f;
        s2 += any2 ? ds2 : 0.0f;
        s2 += any3 ? ds3 : 0.0f;
        c2 += any0 + any1 + any2 + any3;
    }

    float4 v = make_float4(s1, (float)c1, s2, (float)c2);
    v = block_reduce4(v);                 // uniform: every thread participates
    if (threadIdx.x == 0) partials[blockIdx.x] = v;
}

// Pass 2: deterministic fixed-order reduction of the block partials + finalize.
__global__ void __launch_bounds__(BLOCK_THREADS)
heatloss_final(const float4* __restrict__ partials, int n, float* __restrict__ out) {
    float s1 = 0.f, c1 = 0.f, s2 = 0.f, c2 = 0.f;
    for (int i = threadIdx.x; i < n; i += blockDim.x) {
        const float4 p = partials[i];
        s1 += p.x; c1 += p.y; s2 += p.z; c2 += p.w;
    }
    float4 v = make_float4(s1, c1, s2, c2);
    v = block_reduce4(v);
    if (threadIdx.x == 0) {
        const float mean1 = v.x / v.y;
        const float mean2 = v.z / (v.w * (float)NCH);
        out[0] = 0.5f * (mean1 + mean2);
    }
}

extern "C" void kernel_launch(void* const* d_in, const int* in_sizes, int n_in,
                              void* d_out, int out_size, void* d_ws, size_t ws_size,
                              hipStream_t stream) {
    (void)in_sizes; (void)n_in; (void)out_size;
    const float* inp = (const float*)d_in[0];   // input  [32,17,256,256] f32
    const float* tgt = (const float*)d_in[1];   // target [32,17,256,256] f32
    // d_in[2] (masks) and d_in[3] (hull) are unused by the forward pass.
    float* out = (float*)d_out;

    int nblocks = MAIN_BLOCKS;
    const size_t need = (size_t)nblocks * sizeof(float4);
    if (ws_size < need) {                       // defensive fallback only
        nblocks = (int)(ws_size / sizeof(float4));
        if (nblocks < 1) nblocks = 1;
    }
    float4* partials = (float4*)d_ws;

    heatloss_partial<<<nblocks, BLOCK_THREADS, 0, stream>>>(inp, tgt, partials);
    heatloss_final<<<1, BLOCK_THREADS, 0, stream>>>(partials, nblocks, out);
}